// GNNEncoder_25323127177280
// MI455X (gfx1250) — compile-verified
//
#include <hip/hip_runtime.h>

typedef float v2f __attribute__((ext_vector_type(2)));
typedef float v8f __attribute__((ext_vector_type(8)));

#define DFEAT  128
#define PSTR   144   // LDS pair-row stride (in K-pairs); 288 dwords = bank-shift 32

// ---------------------------------------------------------------------------
// zero fill
// ---------------------------------------------------------------------------
__global__ void sage_zero_kernel(float* __restrict__ p, long long n) {
  long long i = (long long)blockIdx.x * blockDim.x + threadIdx.x;
  long long stride = (long long)gridDim.x * blockDim.x;
  for (; i < n; i += stride) p[i] = 0.0f;
}

// ---------------------------------------------------------------------------
// scatter-sum: msg[dst[e], :] += feat[src[e], :]   (+ optional degree count)
// one wave32 per edge; each lane moves 4 floats (float4 gather, L2-resident)
// ---------------------------------------------------------------------------
__global__ void sage_scatter_kernel(const float* __restrict__ feat,
                                    const int* __restrict__ src,
                                    const int* __restrict__ dst,
                                    float* __restrict__ msg,
                                    float* __restrict__ deg,
                                    int E, int addDeg) {
  int e = blockIdx.x * (blockDim.x >> 5) + (threadIdx.x >> 5);
  if (e >= E) return;
  int lane = threadIdx.x & 31;
  int s = src[e];
  int d = dst[e];
  const float4* row = (const float4*)(feat + (size_t)s * DFEAT);
  float4 v = row[lane];
  float* o = msg + (size_t)d * DFEAT + lane * 4;
  atomicAdd(o + 0, v.x);
  atomicAdd(o + 1, v.y);
  atomicAdd(o + 2, v.z);
  atomicAdd(o + 3, v.w);
  if (addDeg && lane == 0) atomicAdd(&deg[d], 1.0f);
}

// ---------------------------------------------------------------------------
// fused SAGE layer GEMM:
//   out = [relu]( (msg/deg) @ Wl + bias + xr @ Wr )
// wave-per-16-row-strip; 8 accumulator tiles of 16x16 f32 over K=128 via
// v_wmma_f32_16x16x4_f32. Weights staged in LDS K-pair-interleaved so each
// lane's B operand is a single aligned ds_load_b64.
// LDS image: sW[p][c][2] with p = k>>1, slot = k&1, pair-row stride PSTR.
// ---------------------------------------------------------------------------
__global__ __launch_bounds__(256) void sage_gemm_kernel(
    const float* __restrict__ msg,   // [N,128] neighbor sums
    const float* __restrict__ deg,   // [N]
    const float* __restrict__ xr,    // [N,128] self features
    const float* __restrict__ Wl,    // [128,128] row-major [k][n]
    const float* __restrict__ bias,  // [128]
    const float* __restrict__ Wr,    // [128,128]
    float* __restrict__ out,         // [N,128]
    int nNodes, int doRelu)
{
  extern __shared__ float smem[];
  float* sWl = smem;                          // 64 * PSTR * 2 floats
  float* sWr = smem + (DFEAT / 2) * PSTR * 2; // 64 * PSTR * 2 floats

  for (int i = threadIdx.x; i < DFEAT * DFEAT; i += blockDim.x) {
    int r = i >> 7;
    int c = i & 127;
    int idx = ((r >> 1) * PSTR + c) * 2 + (r & 1);
    sWl[idx] = Wl[i];
    sWr[idx] = Wr[i];
  }
  __syncthreads();

  int wave = threadIdx.x >> 5;
  int lane = threadIdx.x & 31;
  int tile = blockIdx.x * (blockDim.x >> 5) + wave;
  int mTiles = nNodes >> 4;
  if (tile >= mTiles) return;

  // A-matrix 16x4 f32 layout: lanes 0-15 -> rows, K={0,1}; lanes 16-31 K={2,3}
  int m    = (tile << 4) + (lane & 15);
  int koff = (lane < 16) ? 0 : 2;
  int col  = lane & 15;              // B/C column within an N-tile

  float invd = 1.0f / fmaxf(deg[m], 1.0f);

  v8f acc[8];
  #pragma unroll
  for (int nt = 0; nt < 8; ++nt)
    #pragma unroll
    for (int v = 0; v < 8; ++v) acc[nt][v] = 0.0f;

  const float* aRow = msg + (size_t)m * DFEAT;
  const float* xRow = xr  + (size_t)m * DFEAT;

  // per-lane base into the pair-interleaved LDS image (in floats)
  // element (k, c) pair -> sW[ ((k+koff)>>1)*PSTR*2 + c*2 ]
  int bBase = ((koff >> 1) * PSTR + col) * 2;   // k contributes (k>>1)*PSTR*2

  // GEMM 1: (msg * invd) @ Wl
  for (int k = 0; k < DFEAT; k += 4) {
    v2f a;
    a.x = aRow[k + koff]     * invd;
    a.y = aRow[k + koff + 1] * invd;
    const float* bp = sWl + (k >> 1) * (PSTR * 2) + bBase;
    #pragma unroll
    for (int nt = 0; nt < 8; ++nt) {
      v2f b = *(const v2f*)(bp + nt * 32);      // 16 cols * 2 floats
      acc[nt] = __builtin_amdgcn_wmma_f32_16x16x4_f32(
          false, a, false, b, (short)0, acc[nt], false, false);
    }
  }
  // GEMM 2: xr @ Wr
  for (int k = 0; k < DFEAT; k += 4) {
    v2f a;
    a.x = xRow[k + koff];
    a.y = xRow[k + koff + 1];
    const float* bp = sWr + (k >> 1) * (PSTR * 2) + bBase;
    #pragma unroll
    for (int nt = 0; nt < 8; ++nt) {
      v2f b = *(const v2f*)(bp + nt * 32);
      acc[nt] = __builtin_amdgcn_wmma_f32_16x16x4_f32(
          false, a, false, b, (short)0, acc[nt], false, false);
    }
  }

  // epilogue: bias + optional relu + store
  // C/D layout: VGPR v -> row (v) for lanes 0-15, row (v+8) for lanes 16-31
  int mBase  = tile << 4;
  int rowOff = (lane < 16) ? 0 : 8;
  #pragma unroll
  for (int nt = 0; nt < 8; ++nt) {
    float bv = bias[nt * 16 + col];
    #pragma unroll
    for (int v = 0; v < 8; ++v) {
      float r = acc[nt][v] + bv;
      if (doRelu) r = fmaxf(r, 0.0f);
      out[(size_t)(mBase + rowOff + v) * DFEAT + nt * 16 + col] = r;
    }
  }
}

// ---------------------------------------------------------------------------
// launch
// ---------------------------------------------------------------------------
extern "C" void kernel_launch(void* const* d_in, const int* in_sizes, int n_in,
                              void* d_out, int out_size, void* d_ws, size_t ws_size,
                              hipStream_t stream) {
  const float* x   = (const float*)d_in[0];
  const int*   ei  = (const int*)  d_in[1];
  const float* W1l = (const float*)d_in[2];
  const float* b1  = (const float*)d_in[3];
  const float* W1r = (const float*)d_in[4];
  const float* W2l = (const float*)d_in[5];
  const float* b2  = (const float*)d_in[6];
  const float* W2r = (const float*)d_in[7];
  float* out = (float*)d_out;

  const int E = in_sizes[1] / 2;
  const int n = in_sizes[0] / DFEAT;

  const int* src = ei;
  const int* dst = ei + E;

  float* deg  = (float*)d_ws;                    // [n]
  float* msgb = deg + n;                         // [n,128]
  float* h    = msgb + (size_t)n * DFEAT;        // [n,128]

  const int mTiles      = n >> 4;                // 2500
  const int gemmBlocks  = (mTiles + 7) / 8;      // 8 waves / block
  const int scatBlocks  = (E + 7) / 8;           // 8 edges / block
  const size_t smemB    = (size_t)2 * (DFEAT / 2) * PSTR * 2 * sizeof(float);

  // ---- layer 1 ----
  sage_zero_kernel<<<1024, 256, 0, stream>>>(deg, (long long)n * (DFEAT + 1));
  sage_scatter_kernel<<<scatBlocks, 256, 0, stream>>>(x, src, dst, msgb, deg, E, 1);
  sage_gemm_kernel<<<gemmBlocks, 256, smemB, stream>>>(
      msgb, deg, x, W1l, b1, W1r, h, n, 1);

  // ---- layer 2 ----
  sage_zero_kernel<<<1024, 256, 0, stream>>>(msgb, (long long)n * DFEAT);
  sage_scatter_kernel<<<scatBlocks, 256, 0, stream>>>(h, src, dst, msgb, deg, E, 0);
  sage_gemm_kernel<<<gemmBlocks, 256, smemB, stream>>>(
      msgb, deg, h, W2l, b2, W2r, out, n, 0);
}